// GATEncoder_86981677679216
// MI455X (gfx1250) — compile-verified
//
#include <hip/hip_runtime.h>

#define NEG_SLOPE 0.2f

typedef float v2f __attribute__((ext_vector_type(2)));
typedef float v8f __attribute__((ext_vector_type(8)));

// ---- order-preserving float<->uint encoding for atomicMax-based segment max ----
__device__ __forceinline__ unsigned fenc(float f) {
  unsigned u = __float_as_uint(f);
  return (u & 0x80000000u) ? ~u : (u | 0x80000000u);
}
__device__ __forceinline__ float fdec(unsigned u) {
  return __uint_as_float((u & 0x80000000u) ? (u & 0x7FFFFFFFu) : ~u);
}
#define ENC_NEG_INF 0x007FFFFFu  // fenc(-inf)

// =====================================================================
// H[N x 64] = X[N x K] * W[64 x K]^T   using V_WMMA_F32_16X16X4_F32
// One wave32 per 16x16 output tile; 4 column tiles (64 outputs).
// A (16x4 f32): lanes 0-15 hold row M=lane, K=kk+0/kk+1 in v[0]/v[1];
//               lanes 16-31 hold same rows, K=kk+2/kk+3.  B mirrors A.
// C/D (16x16 f32): vgpr j -> row j (lanes 0-15) / row j+8 (lanes 16-31).
// =====================================================================
__global__ void gemm_xWt_wmma(const float* __restrict__ X,
                              const float* __restrict__ W,
                              float* __restrict__ H,
                              int N, int K) {
  const int lane = threadIdx.x & 31;
  const int wave = blockIdx.x * (blockDim.x >> 5) + (threadIdx.x >> 5);
  const int rowTile = wave >> 2;   // 64/16 = 4 column tiles
  const int colTile = wave & 3;
  if (rowTile * 16 >= N) return;

  const int rc   = lane & 15;        // row (for A) / col (for B) within tile
  const int koff = (lane >> 4) << 1; // 0 or 2
  int arow = rowTile * 16 + rc; if (arow >= N) arow = N - 1; // clamp (guarded store)
  const float* xr = X + (size_t)arow * K + koff;
  const float* wr = W + (size_t)(colTile * 16 + rc) * K + koff;

  v8f acc = {};
  for (int kk = 0; kk < K; kk += 4) {
    v2f a, b;
    a.x = xr[kk];  a.y = xr[kk + 1];
    b.x = wr[kk];  b.y = wr[kk + 1];
    acc = __builtin_amdgcn_wmma_f32_16x16x4_f32(false, a, false, b,
                                                (short)0, acc, false, false);
  }

  const int col  = colTile * 16 + rc;
  const int row0 = rowTile * 16 + ((lane >> 4) << 3);
#pragma unroll
  for (int j = 0; j < 8; ++j) {
    int r = row0 + j;
    if (r < N) H[(size_t)r * 64 + col] = acc[j];
  }
}

// a_src[n,h] = sum_c h[n,h,c]*att_src[h,c] ; same for a_dst
__global__ void att_coef(const float* __restrict__ H,
                         const float* __restrict__ att_s,
                         const float* __restrict__ att_d,
                         float* __restrict__ as_, float* __restrict__ ad_,
                         int N, int heads, int cph) {
  int n = blockIdx.x * blockDim.x + threadIdx.x;
  if (n >= N) return;
  const float* hn = H + (size_t)n * (size_t)(heads * cph);
  for (int h = 0; h < heads; ++h) {
    float ss = 0.f, sd = 0.f;
    for (int c = 0; c < cph; ++c) {
      float v = hn[h * cph + c];
      ss += v * att_s[h * cph + c];
      sd += v * att_d[h * cph + c];
    }
    as_[(size_t)n * heads + h] = ss;
    ad_[(size_t)n * heads + h] = sd;
  }
}

// m = enc(-inf), s = 0, out[n*64+c] = bias[c]   (launched with N*64 threads)
__global__ void init_accum(unsigned* __restrict__ m, float* __restrict__ s,
                           float* __restrict__ out, const float* __restrict__ bias,
                           int N, int heads) {
  long long i = (long long)blockIdx.x * blockDim.x + threadIdx.x;
  long long nh = (long long)N * heads;
  if (i < nh) { m[i] = ENC_NEG_INF; s[i] = 0.f; }
  long long nc = (long long)N * 64;
  if (i < nc) out[i] = bias[i & 63];
}

// Pass 1: e = leaky_relu(a_src[src]+a_dst[dst]); store e; atomicMax m[dst]
template <int HEADS>
__global__ void edge_max(const int* __restrict__ srcA, const int* __restrict__ dstA,
                         const float* __restrict__ as_, const float* __restrict__ ad_,
                         float* __restrict__ ebuf, unsigned* __restrict__ m,
                         long long E, int N) {
  long long t = (long long)blockIdx.x * blockDim.x + threadIdx.x;
  long long total = (E + N) * HEADS;
  if (t >= total) return;
  long long e = t / HEADS;
  int h = (int)(t - e * HEADS);
  int s, d;
  if (e < E) { s = srcA[e]; d = dstA[e]; } else { s = d = (int)(e - E); }
  float v = as_[(size_t)s * HEADS + h] + ad_[(size_t)d * HEADS + h];
  v = (v > 0.f) ? v : NEG_SLOPE * v;
  ebuf[t] = v;
  atomicMax(&m[(size_t)d * HEADS + h], fenc(v));
}

// Pass 2: w = exp(e - m[dst]); overwrite ebuf with w; atomicAdd s[dst]
template <int HEADS>
__global__ void edge_sum(const int* __restrict__ dstA,
                         float* __restrict__ ebuf, const unsigned* __restrict__ m,
                         float* __restrict__ s, long long E, int N) {
  long long t = (long long)blockIdx.x * blockDim.x + threadIdx.x;
  long long total = (E + N) * HEADS;
  if (t >= total) return;
  long long e = t / HEADS;
  int h = (int)(t - e * HEADS);
  int d = (e < E) ? dstA[e] : (int)(e - E);
  float w = __expf(ebuf[t] - fdec(m[(size_t)d * HEADS + h]));
  ebuf[t] = w;
  atomicAdd(&s[(size_t)d * HEADS + h], w);
}

// Pass 3: out[dst,c] += h[src,c] * (w / s[dst,head(c)]) ; one wave32 per edge,
// each lane handles channels {lane, lane+32} -> fully coalesced 64-ch gather/scatter.
template <int HEADS>
__global__ void edge_aggr(const int* __restrict__ srcA, const int* __restrict__ dstA,
                          const float* __restrict__ ebuf, const float* __restrict__ s,
                          const float* __restrict__ H, float* __restrict__ out,
                          long long E, int N) {
  const int lane = threadIdx.x & 31;
  long long e = ((long long)blockIdx.x * blockDim.x + threadIdx.x) >> 5;
  long long total = E + N;
  if (e >= total) return;
  int si, d;
  if (e < E) { si = srcA[e]; d = dstA[e]; } else { si = d = (int)(e - E); }
  constexpr int CPH = 64 / HEADS;
#pragma unroll
  for (int r = 0; r < 2; ++r) {
    int c = lane + r * 32;
    int h = c / CPH;
    float alpha = ebuf[e * HEADS + h] / s[(size_t)d * HEADS + h];
    float msg = H[(size_t)si * 64 + c] * alpha;
    atomicAdd(&out[(size_t)d * 64 + c], msg);
  }
}

__global__ void elu_k(const float* __restrict__ in, float* __restrict__ outp,
                      long long n) {
  long long i = (long long)blockIdx.x * blockDim.x + threadIdx.x;
  if (i >= n) return;
  float v = in[i];
  outp[i] = (v > 0.f) ? v : (__expf(v) - 1.0f);
}

// =====================================================================
extern "C" void kernel_launch(void* const* d_in, const int* in_sizes, int n_in,
                              void* d_out, int out_size, void* d_ws, size_t ws_size,
                              hipStream_t stream) {
  const float* x   = (const float*)d_in[0];
  const int*  eidx = (const int*) d_in[1];   // edge_index [2,E]
  const float* W1  = (const float*)d_in[2];
  const float* b1  = (const float*)d_in[3];
  const float* as1 = (const float*)d_in[4];
  const float* ad1 = (const float*)d_in[5];
  const float* W2  = (const float*)d_in[6];
  const float* b2  = (const float*)d_in[7];
  const float* as2 = (const float*)d_in[8];
  const float* ad2 = (const float*)d_in[9];

  const int F_in = 256;
  const int N = in_sizes[0] / F_in;
  const long long E = in_sizes[1] / 2;
  const int* srcA = eidx;
  const int* dstA = eidx + E;
  float* out = (float*)d_out;

  // ---- workspace carve-up ----
  float* p = (float*)d_ws;
  float* h1   = p; p += (size_t)N * 64;      // layer-1 features (reused as h2)
  float* x2   = p; p += (size_t)N * 64;      // elu(layer-1 output)
  float* out1 = p; p += (size_t)N * 64;      // layer-1 accumulator
  float* a1s  = p; p += (size_t)N * 8;
  float* a1d  = p; p += (size_t)N * 8;
  unsigned* m1 = (unsigned*)p; p += (size_t)N * 8;
  float* s1   = p; p += (size_t)N * 8;
  float* a2s  = p; p += N;
  float* a2d  = p; p += N;
  unsigned* m2 = (unsigned*)p; p += N;
  float* s2   = p; p += N;
  float* ebuf = p;                            // (E+N)*8 floats, reused by layer 2
  float* h2   = h1;                           // h1 dead after elu

  const int B = 256;
  auto cdiv = [](long long a, long long b) { return (unsigned)((a + b - 1) / b); };

  // ================= Layer 1 (heads=8, out=8, concat) =================
  {
    long long waves = (long long)((N + 15) / 16) * 4;
    gemm_xWt_wmma<<<cdiv(waves, 8), B, 0, stream>>>(x, W1, h1, N, 256);
    att_coef<<<cdiv(N, B), B, 0, stream>>>(h1, as1, ad1, a1s, a1d, N, 8, 8);
    init_accum<<<cdiv((long long)N * 64, B), B, 0, stream>>>(m1, s1, out1, b1, N, 8);
    long long tot = (E + N) * 8;
    edge_max<8><<<cdiv(tot, B), B, 0, stream>>>(srcA, dstA, a1s, a1d, ebuf, m1, E, N);
    edge_sum<8><<<cdiv(tot, B), B, 0, stream>>>(dstA, ebuf, m1, s1, E, N);
    edge_aggr<8><<<cdiv((E + N) * 32, B), B, 0, stream>>>(srcA, dstA, ebuf, s1, h1, out1, E, N);
    elu_k<<<cdiv((long long)N * 64, B), B, 0, stream>>>(out1, x2, (long long)N * 64);
  }

  // ================= Layer 2 (heads=1, out=64, mean==identity) =================
  {
    long long waves = (long long)((N + 15) / 16) * 4;
    gemm_xWt_wmma<<<cdiv(waves, 8), B, 0, stream>>>(x2, W2, h2, N, 64);
    att_coef<<<cdiv(N, B), B, 0, stream>>>(h2, as2, ad2, a2s, a2d, N, 1, 64);
    init_accum<<<cdiv((long long)N * 64, B), B, 0, stream>>>(m2, s2, out, b2, N, 1);
    long long tot = (E + N);
    edge_max<1><<<cdiv(tot, B), B, 0, stream>>>(srcA, dstA, a2s, a2d, ebuf, m2, E, N);
    edge_sum<1><<<cdiv(tot, B), B, 0, stream>>>(dstA, ebuf, m2, s2, E, N);
    edge_aggr<1><<<cdiv((E + N) * 32, B), B, 0, stream>>>(srcA, dstA, ebuf, s2, h2, out, E, N);
  }
}